// MDEmbedding_28355374088890
// MI455X (gfx1250) — compile-verified
//
#include <hip/hip_runtime.h>

// CDNA5 / gfx1250 wave32 WMMA types
typedef __attribute__((ext_vector_type(2))) float v2f;
typedef __attribute__((ext_vector_type(8))) float v8f;

namespace {
constexpr int TILE_M = 16;      // rows per wave tile
constexpr int NBASE  = 64;      // output dim
constexpr int NT     = 4;       // 64 / 16 column tiles
constexpr int K1     = 32;      // block-1 embedding dim
constexpr int K2     = 16;      // block-2 embedding dim
constexpr int KT1    = K1 / 4;  // 16x16x4 K-steps for block 1
constexpr int KT2    = K2 / 4;  // 16x16x4 K-steps for block 2
constexpr int OFF1   = 100000;
constexpr int OFF2   = 500000;
}

template <bool FULL>
__device__ __forceinline__ void process_tile(
    int base, int batch, int nl, int hi, int bk,
    const int*   __restrict__ indices,
    const float* __restrict__ t0,
    const float* __restrict__ t1,
    const float* __restrict__ t2,
    const v2f   (&B1)[KT1][NT],
    const v2f   (&B2)[KT2][NT],
    const float (&b1v)[NT],
    const float (&b2v)[NT],
    float*       __restrict__ out)
{
    const int last = batch - 1;

    // ---- A-matrix row owned by this lane (row nl of the tile) ----
    const int ga  = FULL ? (base + nl) : min(base + nl, last);
    const int gi  = indices[ga];
    const int blk = (gi >= OFF1) + (gi >= OFF2);
    const bool is1 = (blk == 1);
    const bool is2 = (blk == 2);

    // Unconditional gathers from a safe row (OOV row 0 when masked: exactly
    // the row the reference touches for non-matching indices; stays hot in
    // WGP$). Mask afterwards with v_cndmask -- no exec-mask branching.
    const int row1 = is1 ? (gi - OFF1 + 1) : 0;
    const int row2 = is2 ? (gi - OFF2 + 1) : 0;
    const v2f* rp1 = (const v2f*)(t1 + row1 * K1) + hi;  // b64 loads, 8B aligned
    const v2f* rp2 = (const v2f*)(t2 + row2 * K2) + hi;

    v2f A1[KT1];
#pragma unroll
    for (int kt = 0; kt < KT1; ++kt) {
        const v2f t = rp1[kt * 2];
        A1[kt].x = is1 ? t.x : 0.0f;
        A1[kt].y = is1 ? t.y : 0.0f;
    }
    v2f A2[KT2];
#pragma unroll
    for (int kt = 0; kt < KT2; ++kt) {
        const v2f t = rp2[kt * 2];
        A2[kt].x = is2 ? t.x : 0.0f;
        A2[kt].y = is2 ? t.y : 0.0f;
    }

    // ---- Accumulators in WMMA C layout:
    //      acc[nt][r] = out[base + r + hi*8][nt*16 + nl]
    // Seed with the block-0 passthrough row (gather) or the masked bias,
    // again with unconditional loads + selects.
    v8f acc[NT];
#pragma unroll
    for (int r = 0; r < 8; ++r) {
        const int mc   = r + hi * 8;
        const int gc   = FULL ? (base + mc) : min(base + mc, last);
        const int gir  = indices[gc];
        const int blkr = (gir >= OFF1) + (gir >= OFF2);
        const bool r0  = (blkr == 0);
        const int row0 = r0 ? (gir + 1) : 0;     // OOV row when masked
        const float* rp0 = t0 + row0 * NBASE + nl;
#pragma unroll
        for (int nt = 0; nt < NT; ++nt) {
            const float g    = rp0[nt * 16];
            const float bias = (blkr == 1) ? b1v[nt] : b2v[nt];
            acc[nt][r] = r0 ? g : bias;
        }
    }

    // ---- out += emb1 @ W1 : K=32 -> 8 K-steps x 4 column tiles ----
#pragma unroll
    for (int kt = 0; kt < KT1; ++kt) {
#pragma unroll
        for (int nt = 0; nt < NT; ++nt) {
            acc[nt] = __builtin_amdgcn_wmma_f32_16x16x4_f32(
                false, A1[kt], false, B1[kt][nt], (short)0, acc[nt], false, false);
        }
    }
    // ---- out += emb2 @ W2 : K=16 -> 4 K-steps x 4 column tiles ----
#pragma unroll
    for (int kt = 0; kt < KT2; ++kt) {
#pragma unroll
        for (int nt = 0; nt < NT; ++nt) {
            acc[nt] = __builtin_amdgcn_wmma_f32_16x16x4_f32(
                false, A2[kt], false, B2[kt][nt], (short)0, acc[nt], false, false);
        }
    }

    // ---- Store the 16x64 tile, streaming (output never re-read) ----
#pragma unroll
    for (int r = 0; r < 8; ++r) {
        const int mc = r + hi * 8;
        if (FULL || (base + mc < batch)) {
            float* op = out + (size_t)(base + mc) * NBASE + nl;
#pragma unroll
            for (int nt = 0; nt < NT; ++nt)
                __builtin_nontemporal_store(acc[nt][r], op + nt * 16);
        }
    }
}

__global__ __launch_bounds__(256) void mdembed_wmma_kernel(
    const int*   __restrict__ indices,
    const float* __restrict__ t0,
    const float* __restrict__ t1,
    const float* __restrict__ t2,
    const float* __restrict__ W1,
    const float* __restrict__ b1,
    const float* __restrict__ W2,
    const float* __restrict__ b2,
    float*       __restrict__ out,
    int batch)
{
    const int lane = threadIdx.x & 31;
    const int nl   = lane & 15;   // col (B/C) or row (A) within 16
    const int hi   = lane >> 4;   // half-wave select
    const int bk   = hi * 2;      // base K within a 4-wide K-step

    const int wavesPerBlk = blockDim.x >> 5;
    const int wave   = blockIdx.x * wavesPerBlk + (threadIdx.x >> 5);
    const int nwave  = gridDim.x * wavesPerBlk;
    const int ntiles = (batch + TILE_M - 1) / TILE_M;

    // ---- One-time per wave: stage W1/W2 into WMMA B-fragment registers ----
    // B (KxN) fragment for 16x16x4: lane L holds N = L&15, K = (L>>4)*2 + {0,1}.
    v2f B1[KT1][NT];
#pragma unroll
    for (int kt = 0; kt < KT1; ++kt) {
        const int k0 = kt * 4 + bk;
#pragma unroll
        for (int nt = 0; nt < NT; ++nt) {
            const int n = nt * 16 + nl;
            B1[kt][nt].x = W1[(k0    ) * NBASE + n];
            B1[kt][nt].y = W1[(k0 + 1) * NBASE + n];
        }
    }
    v2f B2[KT2][NT];
#pragma unroll
    for (int kt = 0; kt < KT2; ++kt) {
        const int k0 = kt * 4 + bk;
#pragma unroll
        for (int nt = 0; nt < NT; ++nt) {
            const int n = nt * 16 + nl;
            B2[kt][nt].x = W2[(k0    ) * NBASE + n];
            B2[kt][nt].y = W2[(k0 + 1) * NBASE + n];
        }
    }
    float b1v[NT], b2v[NT];
#pragma unroll
    for (int nt = 0; nt < NT; ++nt) {
        b1v[nt] = b1[nt * 16 + nl];
        b2v[nt] = b2[nt * 16 + nl];
    }

    // ---- Grid-stride over 16-row tiles; tail handled by a uniform branch ----
    for (int tile = wave; tile < ntiles; tile += nwave) {
        const int base = tile * TILE_M;
        if (base + TILE_M <= batch) {
            process_tile<true >(base, batch, nl, hi, bk, indices, t0, t1, t2,
                                B1, B2, b1v, b2v, out);
        } else {
            process_tile<false>(base, batch, nl, hi, bk, indices, t0, t1, t2,
                                B1, B2, b1v, b2v, out);
        }
    }
}

extern "C" void kernel_launch(void* const* d_in, const int* in_sizes, int n_in,
                              void* d_out, int out_size, void* d_ws, size_t ws_size,
                              hipStream_t stream) {
    const int*   indices = (const int*)  d_in[0];
    const float* t0      = (const float*)d_in[1];
    const float* t1      = (const float*)d_in[2];
    const float* t2      = (const float*)d_in[3];
    const float* W1      = (const float*)d_in[4];
    const float* b1      = (const float*)d_in[5];
    const float* W2      = (const float*)d_in[6];
    const float* b2      = (const float*)d_in[7];
    float*       out     = (float*)d_out;

    const int batch       = in_sizes[0];
    const int threads     = 256;                 // 8 wave32s per workgroup
    const int wavesPerBlk = threads / 32;
    const int ntiles      = (batch + TILE_M - 1) / TILE_M;

    // Grid-stride: cap blocks so the per-wave W-fragment staging amortizes
    // over several tiles while still saturating the WGPs.
    int blocks = (ntiles + wavesPerBlk - 1) / wavesPerBlk;
    if (blocks > 1024) blocks = 1024;
    if (blocks < 1) blocks = 1;

    hipLaunchKernelGGL(mdembed_wmma_kernel, dim3(blocks), dim3(threads), 0, stream,
                       indices, t0, t1, t2, W1, b1, W2, b2, out, batch);
}